// ClassifierLoss_62457414418761
// MI455X (gfx1250) — compile-verified
//
#include <hip/hip_runtime.h>
#include <cstdint>
#include <cstddef>

// Problem constants (from reference setup_inputs).
#define BATCH 64
#define NROI  16384
#define NCLS  21
#define NEGPOS 3

#define K1_ROIS 256                          // rois per block == threads per block
#define K1_BPR  (NROI / K1_ROIS)             // 64 blocks per batch row
#define K2_THREADS 1024

// Low 32 bits of a generic pointer to LDS == byte offset from wave LDS base
// (aperture layout: generic LDS addr = {SHARED_BASE, offset[31:0]}).
__device__ __forceinline__ unsigned lds_off_u32(const void* p) {
    return (unsigned)(size_t)p;
}

// ---------------------------------------------------------------------------
// Kernel 1: per-roi loss_c = lse - picked (0 for positives), per-block partial
// positive count/sum. Prediction rows (21 floats = 84 B) are staged into LDS
// with CDNA5 async global->LDS b128 copies so all HBM traffic is fully
// coalesced 16-byte transfers; per-roi reads then hit LDS conflict-free
// (stride 21 dwords is odd -> injective mod 64 banks).
// ---------------------------------------------------------------------------
__global__ __launch_bounds__(K1_ROIS) void k1_lossc(
    const float* __restrict__ pred, const int* __restrict__ labels,
    float* __restrict__ lossc,
    float* __restrict__ ppos_sum, int* __restrict__ ppos_cnt)
{
    __shared__ __align__(16) float tile[K1_ROIS * NCLS];   // 21504 B
    __shared__ float red_f[K1_ROIS];
    __shared__ int   red_i[K1_ROIS];

    const int tid = threadIdx.x;
    const int b   = blockIdx.y;
    const int n0  = blockIdx.x * K1_ROIS;
    const size_t roi0 = (size_t)b * NROI + (size_t)n0;
    const float* gbase = pred + roi0 * NCLS;               // 16B-aligned: 256*21*4 = 21504 = 16*1344

    // ---- async global -> LDS staging (CDNA5 path, tracked by ASYNCcnt) ----
    {
        const unsigned ldsbase = lds_off_u32(tile);
        const unsigned long long g0 = (unsigned long long)(size_t)gbase;
        const int n128 = (K1_ROIS * NCLS) / 4;             // 1344 16-byte chunks
        for (int i = tid; i < n128; i += K1_ROIS) {
            unsigned l = ldsbase + (unsigned)i * 16u;
            unsigned long long g = g0 + (unsigned long long)i * 16ull;
            asm volatile("global_load_async_to_lds_b128 %0, %1, off"
                         :: "v"(l), "v"(g) : "memory");
        }
#if __has_builtin(__builtin_amdgcn_s_wait_asynccnt)
        __builtin_amdgcn_s_wait_asynccnt(0);
#else
        asm volatile("s_wait_asynccnt 0x0" ::: "memory");
#endif
    }
    __syncthreads();

    // ---- per-roi logsumexp over 21 classes ----
    const float* row = tile + tid * NCLS;
    float v[NCLS];
#pragma unroll
    for (int c = 0; c < NCLS; ++c) v[c] = row[c];

    float m = v[0];
#pragma unroll
    for (int c = 1; c < NCLS; ++c) m = fmaxf(m, v[c]);
    float s = 0.0f;
#pragma unroll
    for (int c = 0; c < NCLS; ++c) s += __expf(v[c] - m);
    const float lse = m + __logf(s);

    const int lab = labels[roi0 + tid];
    const float picked = row[lab];                 // dynamic LDS read, no scratch
    const float ce  = lse - picked;                // >= 0 always
    const bool  pos = (lab > 0);

    lossc[roi0 + tid] = pos ? 0.0f : ce;

    // ---- deterministic block reduction of positive count/sum ----
    red_f[tid] = pos ? ce : 0.0f;
    red_i[tid] = pos ? 1 : 0;
    __syncthreads();
    for (int off = K1_ROIS / 2; off > 0; off >>= 1) {
        if (tid < off) {
            red_f[tid] += red_f[tid + off];
            red_i[tid] += red_i[tid + off];
        }
        __syncthreads();
    }
    if (tid == 0) {
        ppos_sum[b * K1_BPR + blockIdx.x] = red_f[0];
        ppos_cnt[b * K1_BPR + blockIdx.x] = red_i[0];
    }
}

// ---------------------------------------------------------------------------
// Kernel 2: one block per batch row. Reduce partials -> (num_pos, pos_sum),
// K = min(3*num_pos, N-1), 4-pass radix select on float bit patterns
// (valid: all loss_c >= 0) to find the K-th largest value T exactly, then
// row_sum = pos_sum + sum_{x > T} x + ties_taken * T.
// ---------------------------------------------------------------------------
__global__ __launch_bounds__(K2_THREADS) void k2_select(
    const float* __restrict__ lossc,
    const float* __restrict__ ppos_sum, const int* __restrict__ ppos_cnt,
    float* __restrict__ row_sum, int* __restrict__ row_npos)
{
    __shared__ unsigned hist[256];
    __shared__ float red_f[K2_THREADS];
    __shared__ int   red_i[K1_BPR];
    __shared__ unsigned s_bin, s_krem;

    const int tid = threadIdx.x;
    const int b   = blockIdx.x;
    const float* rl = lossc + (size_t)b * NROI;

    // ---- reduce 64 per-block partials (fixed-order tree: deterministic) ----
    if (tid < K1_BPR) {
        red_f[tid] = ppos_sum[b * K1_BPR + tid];
        red_i[tid] = ppos_cnt[b * K1_BPR + tid];
    }
    __syncthreads();
    for (int off = K1_BPR / 2; off > 0; off >>= 1) {
        if (tid < off) {
            red_f[tid] += red_f[tid + off];
            red_i[tid] += red_i[tid + off];
        }
        __syncthreads();
    }
    const float pos_sum = red_f[0];
    const int   num_pos = red_i[0];
    __syncthreads();

    int K = NEGPOS * num_pos;
    if (K > NROI - 1) K = NROI - 1;

    float negsum = 0.0f;
    if (K > 0) {
        unsigned prefix = 0u, mask = 0u;
        unsigned krem = (unsigned)K;
        for (int shift = 24; shift >= 0; shift -= 8) {
            if (tid < 256) hist[tid] = 0u;
            __syncthreads();
            for (int i = tid; i < NROI; i += K2_THREADS) {
                unsigned u = __float_as_uint(rl[i]);
                if ((u & mask) == prefix)
                    atomicAdd(&hist[(u >> shift) & 255u], 1u);   // integer LDS atomic: order-free
            }
            __syncthreads();
            if (tid == 0) {
                unsigned cum = 0; int j = 255;
                for (; j > 0; --j) {
                    unsigned h = hist[j];
                    if (cum + h >= krem) break;
                    cum += h;
                }
                s_bin  = (unsigned)j;
                s_krem = krem - cum;
            }
            __syncthreads();
            prefix |= (s_bin << shift);
            mask   |= (0xFFu << shift);
            krem    = s_krem;
        }
        // sum of all elements strictly greater than threshold T (deterministic tree)
        float local = 0.0f;
        for (int i = tid; i < NROI; i += K2_THREADS) {
            float x = rl[i];
            if (__float_as_uint(x) > prefix) local += x;
        }
        red_f[tid] = local;
        __syncthreads();
        for (int off = K2_THREADS / 2; off > 0; off >>= 1) {
            if (tid < off) red_f[tid] += red_f[tid + off];
            __syncthreads();
        }
        negsum = red_f[0] + (float)krem * __uint_as_float(prefix);
    }

    if (tid == 0) {
        row_sum[b]  = pos_sum + negsum;
        row_npos[b] = num_pos;
    }
}

// ---------------------------------------------------------------------------
// Kernel 3: final scalar = (sum of row sums) / (total positives).
// ---------------------------------------------------------------------------
__global__ __launch_bounds__(64) void k3_final(
    const float* __restrict__ row_sum, const int* __restrict__ row_npos,
    float* __restrict__ out)
{
    __shared__ float rf[BATCH];
    __shared__ int   ri[BATCH];
    const int t = threadIdx.x;
    rf[t] = row_sum[t];
    ri[t] = row_npos[t];
    __syncthreads();
    for (int off = BATCH / 2; off > 0; off >>= 1) {
        if (t < off) { rf[t] += rf[t + off]; ri[t] += ri[t + off]; }
        __syncthreads();
    }
    if (t == 0) out[0] = rf[0] / (float)ri[0];
}

// ---------------------------------------------------------------------------
extern "C" void kernel_launch(void* const* d_in, const int* in_sizes, int n_in,
                              void* d_out, int out_size, void* d_ws, size_t ws_size,
                              hipStream_t stream)
{
    const float* pred   = (const float*)d_in[0];   // [B, N, C] f32
    const int*   labels = (const int*)d_in[1];     // [B, N] i32

    char* ws = (char*)d_ws;
    size_t off = 0;
    float* lossc    = (float*)(ws + off); off += sizeof(float) * (size_t)BATCH * NROI;
    float* ppos_sum = (float*)(ws + off); off += sizeof(float) * BATCH * K1_BPR;
    int*   ppos_cnt = (int*)  (ws + off); off += sizeof(int)   * BATCH * K1_BPR;
    float* row_sum  = (float*)(ws + off); off += sizeof(float) * BATCH;
    int*   row_npos = (int*)  (ws + off); off += sizeof(int)   * BATCH;
    (void)ws_size; (void)in_sizes; (void)n_in; (void)out_size;

    dim3 g1(K1_BPR, BATCH);
    k1_lossc<<<g1, K1_ROIS, 0, stream>>>(pred, labels, lossc, ppos_sum, ppos_cnt);
    k2_select<<<BATCH, K2_THREADS, 0, stream>>>(lossc, ppos_sum, ppos_cnt, row_sum, row_npos);
    k3_final<<<1, 64, 0, stream>>>(row_sum, row_npos, (float*)d_out);
}